// MappedAvgPool_58463094833209
// MI455X (gfx1250) — compile-verified
//
#include <hip/hip_runtime.h>

// Problem constants (from reference)
#define BB   2
#define CC   32
#define HH   512
#define WW   1024
#define HWF  (HH * WW)      // 524288
#define OHH  256
#define OWW  512
#define KP   16             // K*P taps
#define OWT  32             // ow positions per block

typedef __attribute__((ext_vector_type(2))) float v2f;
typedef __attribute__((ext_vector_type(8))) float v8f;

// ---------------------------------------------------------------------------
// Kernel 1: transpose x (B, C, HW) -> xt (B, HW, C) so gathers over spatial
// index become 128B-contiguous across the 32 channels (coalesced wave loads).
// ---------------------------------------------------------------------------
__global__ __launch_bounds__(256) void xpose_kernel(const float* __restrict__ x,
                                                    float* __restrict__ xt) {
    __shared__ float tile[32][33];
    const int tx = threadIdx.x;          // 0..31 -> spatial within tile / channel
    const int ty = threadIdx.y;          // 0..7
    const int b  = blockIdx.y;
    const long s0 = (long)blockIdx.x * 32;

    const float* xb  = x  + (long)b * CC * HWF;
    float*       xtb = xt + (long)b * HWF * CC;

#pragma unroll
    for (int i = 0; i < 4; ++i) {
        const int c = ty + 8 * i;                    // channel 0..31
        tile[c][tx] = xb[(long)c * HWF + s0 + tx];   // coalesced along spatial
    }
    __syncthreads();
#pragma unroll
    for (int i = 0; i < 4; ++i) {
        const int s = ty + 8 * i;                    // spatial within tile
        xtb[(s0 + s) * CC + tx] = tile[tx][s];       // coalesced along channel
    }
}

// ---------------------------------------------------------------------------
// Kernel 2: gather + weighted 16-tap reduction via V_WMMA_F32_16X16X4_F32.
// Block = 256 threads (8 waves), handles (b, oh, 32 ow positions).
// Wave: 4 positions; per position, two channel-halves of 16 channels each;
// 4 chained WMMAs (K=4 taps each, B = ones) produce per-channel rowsums.
// ---------------------------------------------------------------------------
__global__ __launch_bounds__(256) void mapped_pool_wmma_kernel(
        const float* __restrict__ xt,      // (B, HW, C)
        const int*   __restrict__ smap,    // (OH, OW, K, P, 2)
        const float* __restrict__ wts,     // (OH, OW, K, P)
        float*       __restrict__ out) {   // (B, C, OH, OW)
    __shared__ int   s_lin[OWT][KP];
    __shared__ float s_w[OWT][KP];
    __shared__ float s_res[OWT][CC];       // [pos][c]

    const int tid = threadIdx.x;
    const int bid = blockIdx.x;            // B * OH * (OW/OWT) = 8192
    const int owb = bid & 15;              // OW/OWT = 16 tiles
    const int oh  = (bid >> 4) & 255;
    const int b   = bid >> 12;
    const int ow0 = owb * OWT;

    // Stage tap metadata (uniform across channels) into LDS: 512 entries.
    for (int e = tid; e < OWT * KP; e += 256) {
        const int  pos  = e >> 4;
        const int  t    = e & 15;
        const long base = (long)((oh * OWW + ow0 + pos) * KP + t);
        const int  col  = smap[base * 2 + 0];
        const int  row  = smap[base * 2 + 1];
        s_lin[pos][t] = row * WW + col;
        s_w[pos][t]   = wts[base] * 0.25f;     // fold the /K
    }
    __syncthreads();

    const int lane = tid & 31;
    const int wave = tid >> 5;             // 0..7
    const int half = lane >> 4;            // A-matrix k-slot half (0 or 1)
    const int mc   = lane & 15;            // A-matrix row m = channel-in-block

    const float* xb = xt + (long)b * HWF * CC;
    const v2f ones = {1.0f, 1.0f};         // B = all-ones 4x16 -> rowsum

#pragma unroll
    for (int pi = 0; pi < 4; ++pi) {
        const int pos = wave * 4 + pi;     // uniform per wave
#pragma unroll
        for (int ch = 0; ch < 2; ++ch) {   // channel halves: c0 = 16*ch
            const int c = 16 * ch + mc;
            v8f acc = {};
#pragma unroll
            for (int q = 0; q < 4; ++q) {
                // A layout (16x4 f32): lanes 0-15 hold k={0,1}, lanes 16-31 k={2,3}
                const int t0 = 4 * q + 2 * half;
                const int l0 = s_lin[pos][t0];
                const int l1 = s_lin[pos][t0 + 1];
                const float a0 = xb[l0 * CC + c] * s_w[pos][t0];
                const float a1 = xb[l1 * CC + c] * s_w[pos][t0 + 1];
                v2f a = {a0, a1};
                acc = __builtin_amdgcn_wmma_f32_16x16x4_f32(
                    false, a, false, ones, (short)0, acc, false, false);
            }
            // D layout: lane<16 holds M=0..7 in acc[0..7]; lane>=16 holds M=8..15.
            // All N columns equal (B = ones) -> lanes 0 and 16 carry everything.
            if (mc == 0) {
                const int cbase = 16 * ch + 8 * half;
#pragma unroll
                for (int r = 0; r < 8; ++r) s_res[pos][cbase + r] = acc[r];
            }
        }
    }
    __syncthreads();

    // Coalesced writeback: 32 consecutive ow per channel = 128B runs.
    for (int e = tid; e < OWT * CC; e += 256) {
        const int c   = e >> 5;
        const int pos = e & 31;
        out[(((long)b * CC + c) * OHH + oh) * OWW + ow0 + pos] = s_res[pos][c];
    }
}

// ---------------------------------------------------------------------------
// Fallback (workspace too small for the transposed copy): direct gather.
// ---------------------------------------------------------------------------
__global__ __launch_bounds__(256) void mapped_pool_fallback_kernel(
        const float* __restrict__ x, const int* __restrict__ smap,
        const float* __restrict__ wts, float* __restrict__ out) {
    const long i = (long)blockIdx.x * 256 + threadIdx.x;   // over B*C*OH*OW
    if (i >= (long)BB * CC * OHH * OWW) return;
    const int ow = (int)(i % OWW);
    const int oh = (int)((i / OWW) % OHH);
    const int c  = (int)((i / ((long)OWW * OHH)) % CC);
    const int b  = (int)(i / ((long)OWW * OHH * CC));
    const float* xb   = x + ((long)b * CC + c) * HWF;
    const long   base = (long)(oh * OWW + ow) * KP;
    float acc = 0.0f;
#pragma unroll
    for (int t = 0; t < KP; ++t) {
        const int col = smap[(base + t) * 2 + 0];
        const int row = smap[(base + t) * 2 + 1];
        acc += xb[row * WW + col] * wts[base + t];
    }
    out[i] = acc * 0.25f;
}

// ---------------------------------------------------------------------------
extern "C" void kernel_launch(void* const* d_in, const int* in_sizes, int n_in,
                              void* d_out, int out_size, void* d_ws, size_t ws_size,
                              hipStream_t stream) {
    (void)in_sizes; (void)n_in; (void)out_size;
    const float* x    = (const float*)d_in[0];
    const int*   smap = (const int*)d_in[1];
    const float* wts  = (const float*)d_in[2];
    float*       out  = (float*)d_out;

    const size_t need = (size_t)BB * HWF * CC * sizeof(float);   // 128 MiB
    if (ws_size >= need && d_ws != nullptr) {
        float* xt = (float*)d_ws;
        dim3 tb(32, 8);
        dim3 tg(HWF / 32, BB);                                   // (16384, 2)
        xpose_kernel<<<tg, tb, 0, stream>>>(x, xt);
        const int nblocks = BB * OHH * (OWW / OWT);              // 8192
        mapped_pool_wmma_kernel<<<nblocks, 256, 0, stream>>>(xt, smap, wts, out);
    } else {
        const long total = (long)BB * CC * OHH * OWW;
        mapped_pool_fallback_kernel<<<(int)((total + 255) / 256), 256, 0, stream>>>(
            x, smap, wts, out);
    }
}